// sigpy_nufft_75522704933325
// MI455X (gfx1250) — compile-verified
//
#include <hip/hip_runtime.h>
#include <math.h>

typedef __attribute__((ext_vector_type(2))) float v2f;
typedef __attribute__((ext_vector_type(8))) float v8f;

#define OS   320      // oversampled grid (1.25 * 256)
#define IMS  256      // image size
#define NCH  8        // channels
#define NPTS 65536    // 16 * 4096 trajectory points
#define PADB 32       // (320/2 - 256/2) zero-pad before

__device__ __forceinline__ float kb_beta() {
    // pi * sqrt((W/os*(os-0.5))^2 - 0.8) = pi*sqrt(12.16)
    return 3.14159265358979f * sqrtf(12.16f);
}

// Abramowitz & Stegun 9.8.1 / 9.8.2, |err| < 2e-7 relative
__device__ __forceinline__ float bessel_i0(float x) {
    float ax = fabsf(x);
    if (ax < 3.75f) {
        float t = ax * (1.0f / 3.75f); t *= t;
        return 1.0f + t*(3.5156229f + t*(3.0899424f + t*(1.2067492f +
               t*(0.2659732f + t*(0.0360768f + t*0.0045813f)))));
    }
    float t = 3.75f / ax;
    float p = 0.39894228f + t*(0.01328592f + t*(0.00225319f + t*(-0.00157565f +
              t*(0.00916281f + t*(-0.02057706f + t*(0.02635537f +
              t*(-0.01647633f + t*0.00392377f)))))));
    return expf(ax) * rsqrtf(ax) * p;
}

// Centered DFT matrix with shifts + padding folded in:
// D[k][j] = (-1)^(k+j) * exp(-2*pi*i * k*(j+32)/320),  k in [0,320), j in [0,256)
__global__ void build_dft(float* __restrict__ Dre, float* __restrict__ Dim) {
    int idx = blockIdx.x * blockDim.x + threadIdx.x;
    if (idx >= OS * IMS) return;
    int k = idx / IMS, j = idx % IMS;
    int m = (k * (j + PADB)) % OS;                 // exact integer phase reduction
    float th = 6.28318530717958647692f * (float)m * (1.0f / (float)OS);
    float s = ((k + j) & 1) ? -1.0f : 1.0f;        // (+32 is even)
    Dre[idx] = s * cosf(th);
    Dim[idx] = -s * sinf(th);
}

__device__ __forceinline__ float apod1(int i) {
    float beta = kb_beta();
    float u = 3.14159265358979f * 6.0f * (float)(i - IMS / 2) * (1.0f / (float)OS);
    float a = sqrtf(beta * beta - u * u);          // always > 0 for |i-128|<=128
    return a / sinhf(a);
}

__global__ void apodize(const float* __restrict__ ir, const float* __restrict__ ii,
                        float* __restrict__ Are, float* __restrict__ Aim) {
    int idx = blockIdx.x * blockDim.x + threadIdx.x;
    if (idx >= NCH * IMS * IMS) return;
    int rem = idx % (IMS * IMS);
    int y = rem / IMS, x = rem % IMS;
    float w = apod1(y) * apod1(x) * (1.0f / 256.0f);   // includes 1/sqrt(H*W)
    Are[idx] = ir[idx] * w;
    Aim[idx] = ii[idx] * w;
}

// Complex GEMM C[MxN] = A[MxK] * Bw[KxN] via V_WMMA_F32_16X16X4_F32.
// Planar complex (separate re/im planes). A row-major lda=K.
// BT=true : B stored [N][K] row-major (ldb=K), Bw[k][n] = B[n*ldb+k]   (A*B^T)
// BT=false: B stored [K][N] row-major (ldb=N), Bw[k][n] = B[k*ldb+n]   (A*B)
// One wave (32 lanes) owns one 16x16 output tile; blockDim=(32,4).
template <bool BT>
__global__ void cgemm_wmma(const float* __restrict__ Are, const float* __restrict__ Aim,
                           const float* __restrict__ Bre, const float* __restrict__ Bim,
                           float* __restrict__ Cre, float* __restrict__ Cim,
                           int M, int N, int K, int lda, int ldb, int ldc,
                           long aStride, long bStride, long cStride) {
    int mtiles = M >> 4, ntiles = N >> 4;
    int tile = blockIdx.x * blockDim.y + threadIdx.y;
    if (tile >= mtiles * ntiles) return;           // uniform per-wave exit: EXEC stays all-1s
    int ch = blockIdx.y;
    const float* are = Are + (long)ch * aStride;
    const float* aim = Aim + (long)ch * aStride;
    const float* bre = Bre + (long)ch * bStride;
    const float* bim = Bim + (long)ch * bStride;
    float* cre = Cre + (long)ch * cStride;
    float* cim = Cim + (long)ch * cStride;

    int m0 = (tile / ntiles) << 4;
    int n0 = (tile % ntiles) << 4;
    int lane = threadIdx.x;
    int r = lane & 15;                             // row within tile (A: M row, B: N col)
    int h = lane >> 4;                             // half-wave selects K pair {0,1} vs {2,3}

    v8f accRR = {}, accII = {}, accRI = {}, accIR = {};
    const float* arow  = are + (long)(m0 + r) * lda;
    const float* airow = aim + (long)(m0 + r) * lda;

    for (int k = 0; k < K; k += 4) {
        int ka = k + 2 * h;
        v2f a_re = { arow[ka],  arow[ka + 1]  };
        v2f a_im = { airow[ka], airow[ka + 1] };
        v2f b_re, b_im;
        if (BT) {
            const float* br = bre + (long)(n0 + r) * ldb + ka;
            const float* bi = bim + (long)(n0 + r) * ldb + ka;
            b_re = (v2f){ br[0], br[1] };
            b_im = (v2f){ bi[0], bi[1] };
        } else {
            const float* br = bre + (long)ka * ldb + (n0 + r);
            const float* bi = bim + (long)ka * ldb + (n0 + r);
            b_re = (v2f){ br[0], br[ldb] };
            b_im = (v2f){ bi[0], bi[ldb] };
        }
        accRR = __builtin_amdgcn_wmma_f32_16x16x4_f32(false, a_re, false, b_re, (short)0, accRR, false, false);
        accII = __builtin_amdgcn_wmma_f32_16x16x4_f32(false, a_im, false, b_im, (short)0, accII, false, false);
        accRI = __builtin_amdgcn_wmma_f32_16x16x4_f32(false, a_re, false, b_im, (short)0, accRI, false, false);
        accIR = __builtin_amdgcn_wmma_f32_16x16x4_f32(false, a_im, false, b_re, (short)0, accIR, false, false);
    }

    // C/D layout: VGPR i <-> row m0 + i + 8*h, column n0 + r
    #pragma unroll
    for (int i = 0; i < 8; ++i) {
        long o = (long)(m0 + i + 8 * h) * ldc + (n0 + r);
        cre[o] = accRR[i] - accII[i];
        cim[o] = accRI[i] + accIR[i];
    }
}

// Kaiser-Bessel 7x7 gather-interp per trajectory point, all 8 channels per thread.
__global__ void kb_interp(const float* __restrict__ trj,
                          const float* __restrict__ Gre, const float* __restrict__ Gim,
                          float* __restrict__ out) {
    int p = blockIdx.x * blockDim.x + threadIdx.x;
    if (p >= NPTS) return;
    float beta = kb_beta();
    float cy = trj[2 * p]     * 1.25f + 160.0f;
    float cx = trj[2 * p + 1] * 1.25f + 160.0f;
    float y0 = ceilf(cy - 3.0f);
    float x0 = ceilf(cx - 3.0f);
    float wy[7], wx[7];
    int   iy[7], ix[7];
    #pragma unroll
    for (int t = 0; t < 7; ++t) {
        float py = y0 + (float)t, px = x0 + (float)t;
        float uy = (py - cy) * (1.0f / 3.0f);
        float ux = (px - cx) * (1.0f / 3.0f);
        wy[t] = (fabsf(uy) > 1.0f) ? 0.0f
              : bessel_i0(beta * sqrtf(fmaxf(1.0f - uy * uy, 0.0f)));
        wx[t] = (fabsf(ux) > 1.0f) ? 0.0f
              : bessel_i0(beta * sqrtf(fmaxf(1.0f - ux * ux, 0.0f)));
        int a = (int)py; a %= OS; if (a < 0) a += OS; iy[t] = a;
        int b = (int)px; b %= OS; if (b < 0) b += OS; ix[t] = b;
    }
    for (int c = 0; c < NCH; ++c) {
        const float* gr = Gre + (long)c * (OS * OS);
        const float* gi = Gim + (long)c * (OS * OS);
        float sre = 0.0f, sim = 0.0f;
        #pragma unroll
        for (int ty = 0; ty < 7; ++ty) {
            const float* grr = gr + iy[ty] * OS;
            const float* gii = gi + iy[ty] * OS;
            float wyt = wy[ty];
            #pragma unroll
            for (int tx = 0; tx < 7; ++tx) {
                float w = wyt * wx[tx];
                sre += w * grr[ix[tx]];
                sim += w * gii[ix[tx]];
            }
        }
        // out shape [2][1][8][16][4096] flat
        out[(long)c * NPTS + p]          = sre * (1.0f / 36.0f);
        out[(long)(NCH + c) * NPTS + p]  = sim * (1.0f / 36.0f);
    }
}

extern "C" void kernel_launch(void* const* d_in, const int* in_sizes, int n_in,
                              void* d_out, int out_size, void* d_ws, size_t ws_size,
                              hipStream_t stream) {
    const float* img_real = (const float*)d_in[0];   // [1,8,256,256]
    const float* img_imag = (const float*)d_in[1];   // [1,8,256,256]
    const float* trj      = (const float*)d_in[2];   // [1,16,4096,2]

    float* ws  = (float*)d_ws;
    float* Dre = ws;                    // 320*256        = 81920
    float* Dim = Dre + 81920;
    float* Are = Dim + 81920;           // 8*256*256      = 524288
    float* Aim = Are + 524288;
    float* Tre = Aim + 524288;          // 8*256*320      = 655360
    float* Tim = Tre + 655360;
    float* Gre = Tim + 655360;          // 8*320*320      = 819200
    float* Gim = Gre + 819200;          // total ~16.6 MB of d_ws

    build_dft<<<(OS * IMS + 255) / 256, 256, 0, stream>>>(Dre, Dim);
    apodize<<<(NCH * IMS * IMS + 255) / 256, 256, 0, stream>>>(img_real, img_imag, Are, Aim);

    dim3 blk(32, 4);
    // Pass 1 (transform along x): T[256x320] = A[256x256] * D^T, D stored [320][256]
    int tiles1 = (IMS / 16) * (OS / 16);   // 16*20 = 320
    cgemm_wmma<true><<<dim3((tiles1 + 3) / 4, NCH), blk, 0, stream>>>(
        Are, Aim, Dre, Dim, Tre, Tim,
        IMS, OS, IMS, IMS, IMS, OS,
        (long)IMS * IMS, 0L, (long)IMS * OS);

    // Pass 2 (transform along y): G[320x320] = D[320x256] * T[256x320]
    int tiles2 = (OS / 16) * (OS / 16);    // 20*20 = 400
    cgemm_wmma<false><<<dim3((tiles2 + 3) / 4, NCH), blk, 0, stream>>>(
        Dre, Dim, Tre, Tim, Gre, Gim,
        OS, OS, IMS, IMS, OS, OS,
        0L, (long)IMS * OS, (long)OS * OS);

    kb_interp<<<(NPTS + 127) / 128, 128, 0, stream>>>(trj, Gre, Gim, (float*)d_out);
}